// ROINet_24395414241755
// MI455X (gfx1250) — compile-verified
//
#include <hip/hip_runtime.h>
#include <hip/hip_bf16.h>

// ---------------------------------------------------------------------------
// ROI-Align for MI455X (gfx1250).  Pipeline:
//   1) NCHW -> NHWC transpose into d_ws (coalesced 512B/sample reads after).
//   2) Per (roi, p-row): build Vsum[56][128] in LDS (y-mix + wx folded, f32).
//   3) out[q, c] = block-diag selector (16x56) x Vsum (56x128) via
//      V_WMMA_F32_16X16X4_F32, 14 K-steps, one 16-channel chunk per wave.
// ---------------------------------------------------------------------------

typedef float v2f __attribute__((ext_vector_type(2)));
typedef float v8f __attribute__((ext_vector_type(8)));

#define RA_P    7
#define RA_MG   4
#define RA_C    128
#define RA_H    64
#define RA_W    64
#define VROWS   56          // 7 q * 4 b * 2 x-corners
#define VSTRIDE 144         // 128 + 16 pad -> lane16-31 B-rows hit disjoint banks

// ---------------------------------------------------------------------------
// Kernel 1: NCHW -> NHWC transpose.  One block per (n, y) slab of 128c x 64x.
// ---------------------------------------------------------------------------
__global__ __launch_bounds__(256)
void ROINet_nchw_to_nhwc(const float* __restrict__ in, float* __restrict__ outT)
{
    __shared__ float tile[RA_W][RA_C + 1];   // +1 pad: conflict-free both phases
    const int n   = blockIdx.x >> 6;
    const int y   = blockIdx.x & 63;
    const int tid = threadIdx.x;

    // read phase: consecutive lanes -> consecutive x (coalesced)
    const int xr = tid & 63;
    for (int c = (tid >> 6); c < RA_C; c += 4)
        tile[xr][c] = in[(((size_t)n * RA_C + c) * RA_H + y) * RA_W + xr];
    __syncthreads();

    // write phase: consecutive lanes -> consecutive c (coalesced 512B)
    const int cw = tid & 127;
    for (int x = (tid >> 7); x < RA_W; x += 2)
        outT[(((size_t)n * RA_H + y) * RA_W + x) * RA_C + cw] = tile[x][cw];
}

// ---------------------------------------------------------------------------
// Kernel 2: ROI-Align.  Grid = (R, 7), block = 256 (8 waves).
// NHWC=true  : feat points at transposed d_ws (fast path).
// NHWC=false : direct strided NCHW reads (fallback if ws too small).
// ---------------------------------------------------------------------------
template <bool NHWC>
__global__ __launch_bounds__(256)
void ROINet_roi_align(const float* __restrict__ feat,
                      const float* __restrict__ rois,
                      float* __restrict__ out)
{
    __shared__ float sV[VROWS * VSTRIDE];
    __shared__ int   sYI[8];
    __shared__ float sWY[8];
    __shared__ int   sXI[VROWS];
    __shared__ float sWX[VROWS];
    __shared__ float sInv;
    __shared__ int   sN;

    const int r   = blockIdx.x;
    const int p   = blockIdx.y;
    const int tid = threadIdx.x;

    // ---- ROI parameters (uniform; every thread recomputes, cheap) ----
    const float* roi = rois + (size_t)r * 5;
    const int   n     = (int)roi[0];
    const float x1    = roi[1] * 0.0625f;
    const float y1    = roi[2] * 0.0625f;
    const float x2    = roi[3] * 0.0625f;
    const float y2    = roi[4] * 0.0625f;
    const float roi_w = fmaxf(x2 - x1, 1.0f);
    const float roi_h = fmaxf(y2 - y1, 1.0f);
    const float bin_w = roi_w * (1.0f / RA_P);
    const float bin_h = roi_h * (1.0f / RA_P);
    const float gw    = fminf(fmaxf(ceilf(roi_w * (1.0f / RA_P)), 1.0f), (float)RA_MG);
    const float gh    = fminf(fmaxf(ceilf(roi_h * (1.0f / RA_P)), 1.0f), (float)RA_MG);

    if (tid == 0) { sInv = 1.0f / (gh * gw); sN = n; }

    if (tid < VROWS) {                       // 56 x-slots: j = 8q + 2b + corner
        const int q = tid >> 3, k = tid & 7, b = k >> 1, corner = k & 1;
        const float x  = x1 + q * bin_w + ((float)b + 0.5f) * (bin_w / gw);
        const bool  ok = ((float)b < gw) && (x >= -1.0f) && (x <= (float)RA_W);
        const float cx = fminf(fmaxf(x, 0.0f), (float)(RA_W - 1));
        const int   xl = (int)floorf(cx);
        const int   xh = min(xl + 1, RA_W - 1);
        const float fx = cx - (float)xl;
        sXI[tid] = corner ? xh : xl;
        sWX[tid] = ok ? (corner ? fx : 1.0f - fx) : 0.0f;
    } else if (tid >= 64 && tid < 72) {      // 8 y-terms: t = 2a + corner
        const int t = tid - 64, a = t >> 1, corner = t & 1;
        const float y  = y1 + p * bin_h + ((float)a + 0.5f) * (bin_h / gh);
        const bool  ok = ((float)a < gh) && (y >= -1.0f) && (y <= (float)RA_H);
        const float cy = fminf(fmaxf(y, 0.0f), (float)(RA_H - 1));
        const int   yl = (int)floorf(cy);
        const int   yh = min(yl + 1, RA_H - 1);
        const float fy = cy - (float)yl;
        sYI[t] = corner ? yh : yl;
        sWY[t] = ok ? (corner ? fy : 1.0f - fy) : 0.0f;
    }
    __syncthreads();

    // ---- Step 1: Vsum[j][c] = wx[j] * sum_t wy[t] * feat[y_t, x_j, c] ----
    const int lane = tid & 31;
    const int wv   = tid >> 5;        // wave id 0..7
    const int c4   = lane << 2;       // float4 channel base
    const int nImg = sN;

    for (int j = wv; j < VROWS; j += 8) {
        const int   xi  = sXI[j];
        const float wxj = sWX[j];     // uniform across the wave
        float4 acc = make_float4(0.f, 0.f, 0.f, 0.f);
        if (wxj != 0.0f) {
            #pragma unroll
            for (int t = 0; t < 8; ++t) {
                const float wt = sWY[t];   // uniform
                if (wt != 0.0f) {
                    const int yi = sYI[t];
                    float4 v;
                    if (NHWC) {
                        const float* src = feat +
                            ((((size_t)nImg * RA_H + yi) * RA_W + xi) * RA_C) + c4;
                        v = *(const float4*)src;      // 16B aligned, coalesced
                    } else {
                        const float* src = feat +
                            ((size_t)nImg * RA_C + c4) * (RA_H * RA_W) + yi * RA_W + xi;
                        v.x = src[0];
                        v.y = src[RA_H * RA_W];
                        v.z = src[2 * RA_H * RA_W];
                        v.w = src[3 * RA_H * RA_W];
                    }
                    acc.x += wt * v.x; acc.y += wt * v.y;
                    acc.z += wt * v.z; acc.w += wt * v.w;
                }
            }
            acc.x *= wxj; acc.y *= wxj; acc.z *= wxj; acc.w *= wxj;
        }
        float* dst = &sV[j * VSTRIDE + c4];
        dst[0] = acc.x; dst[1] = acc.y; dst[2] = acc.z; dst[3] = acc.w;
    }
    __syncthreads();

    const float inv = sInv;

#if defined(__HIP_DEVICE_COMPILE__) && __has_builtin(__builtin_amdgcn_wmma_f32_16x16x4_f32)
    // ---- Step 2 (WMMA): out[16(q-pad) x 16c] = A(16x56, 0/1 block-diag) x Vsum ----
    // A layout (ISA 7.12.2, 32-bit A 16x4): M = lane&15; v0 holds K=0 (lanes 0-15)
    // / K=2 (lanes 16-31), v1 holds K=1 / K=3.  B mirrored: N = lane&15.
    {
        const int cb = wv << 4;           // 16-channel chunk per wave: 8*16 = 128
        const int mn = lane & 15;
        const int hi = lane >> 4;         // 0: K+0/K+1, 1: K+2/K+3
        v8f dacc = {0.f, 0.f, 0.f, 0.f, 0.f, 0.f, 0.f, 0.f};
        #pragma unroll
        for (int ks = 0; ks < 14; ++ks) { // K = 56 in steps of 4
            const int k0 = ks * 4 + (hi << 1);
            const int k1 = k0 + 1;
            v2f A, B;
            A.x = ((k0 >> 3) == mn) ? 1.0f : 0.0f;   // selector: row q owns k in [8q,8q+8)
            A.y = ((k1 >> 3) == mn) ? 1.0f : 0.0f;
            B.x = sV[k0 * VSTRIDE + cb + mn];
            B.y = sV[k1 * VSTRIDE + cb + mn];
            dacc = __builtin_amdgcn_wmma_f32_16x16x4_f32(
                false, A, false, B, (short)0, dacc, false, false);
        }
        // D layout: lanes 0-15 -> N = lane, VGPR i -> M = i.  Only M = q in [0,7)
        // is live (lanes 16-31 hold M >= 8, all zero by construction of A).
        if (hi == 0) {
            const int c = cb + mn;
            float* o = out + (((size_t)r * RA_C + c) * RA_P + p) * RA_P;
            #pragma unroll
            for (int q = 0; q < RA_P; ++q) o[q] = dacc[q] * inv;
        }
    }
#else
    // ---- Step 2 (VALU fallback): out[q,c] = sum_{k=8q..8q+7} Vsum[k][c] ----
    {
        const int q = tid >> 5;           // waves 0..6 active, wave 7 idle
        if (q < RA_P) {
            const int cb4 = (tid & 31) << 2;
            float4 acc = make_float4(0.f, 0.f, 0.f, 0.f);
            #pragma unroll
            for (int k = 0; k < 8; ++k) {
                const float* vr = &sV[(8 * q + k) * VSTRIDE + cb4];
                acc.x += vr[0]; acc.y += vr[1]; acc.z += vr[2]; acc.w += vr[3];
            }
            const size_t base = (((size_t)r * RA_C + cb4) * RA_P + p) * RA_P + q;
            out[base          ] = acc.x * inv;
            out[base +  49    ] = acc.y * inv;
            out[base +  98    ] = acc.z * inv;
            out[base + 147    ] = acc.w * inv;
        }
    }
#endif
}

// ---------------------------------------------------------------------------
extern "C" void kernel_launch(void* const* d_in, const int* in_sizes, int n_in,
                              void* d_out, int out_size, void* d_ws, size_t ws_size,
                              hipStream_t stream)
{
    const float* feat  = (const float*)d_in[0];   // (B,T,C,H,W) -> (32,128,64,64)
    const float* tubes = (const float*)d_in[1];   // (NT,T,5)    -> (768,5)
    float*       outp  = (float*)d_out;

    const int R    = in_sizes[1] / 5;                               // 768
    const int NImg = in_sizes[0] / (RA_C * RA_H * RA_W);            // 32
    const size_t needT = (size_t)NImg * RA_H * RA_W * RA_C * sizeof(float);

    dim3 grid((unsigned)R, RA_P);
    if (ws_size >= needT) {
        ROINet_nchw_to_nhwc<<<NImg * RA_H, 256, 0, stream>>>(feat, (float*)d_ws);
        ROINet_roi_align<true><<<grid, 256, 0, stream>>>((const float*)d_ws, tubes, outp);
    } else {
        ROINet_roi_align<false><<<grid, 256, 0, stream>>>(feat, tubes, outp);
    }
}